// IRLSStepModule_2783138808065
// MI455X (gfx1250) — compile-verified
//
#include <hip/hip_runtime.h>
#include <hip/hip_bf16.h>

typedef __attribute__((ext_vector_type(2))) float v2f;
typedef __attribute__((ext_vector_type(8))) float v8f;

#define WEIGHT_DELTA 1e-3f
#define EIGH_EPS     1e-6f
#define JACOBI_SWEEPS 10

// 16-lane butterfly add using DPP16 (no LDS traffic):
//   xor1 = quad_perm{1,0,3,2} (0xB1), xor2 = quad_perm{2,3,0,1} (0x4E),
//   xor4 = row_half_mirror (0x141),   xor8 = row_mirror (0x140)
template <int CTRL>
__device__ __forceinline__ float add_dpp(float x) {
    int y = __builtin_amdgcn_update_dpp(0, __float_as_int(x), CTRL,
                                        /*row_mask=*/0xF, /*bank_mask=*/0xF,
                                        /*bound_ctrl=*/true);
    return x + __int_as_float(y);
}
__device__ __forceinline__ float half_reduce16(float x) {
    x = add_dpp<0x0B1>(x);
    x = add_dpp<0x04E>(x);
    x = add_dpp<0x141>(x);
    x = add_dpp<0x140>(x);
    return x;
}

// One wave (32 lanes) per batch. Lane layout matches V_WMMA_F32_16X16X4_F32:
//   c = lane & 15  -> matrix column (valid for c < 9)
//   h = lane >> 4  -> half: VGPR j of A/B operand holds K = 2h + j
// Per step we consume 4 rows of G = diag(sqrt(w)) * A and do a symmetric
// rank-4 update M += G_chunk^T G_chunk on the matrix core (A-op == B-op regs).
__global__ __launch_bounds__(256) void irls_step_kernel(
    const float* __restrict__ coords,  // [B, N, 9]
    const float* __restrict__ fest,    // [B, 9]
    float* __restrict__ out,           // [B, 9]
    int Bn, int N)
{
    __shared__ float shM[8][81];
    __shared__ float shV[8][81];

    const int tid  = threadIdx.x;
    const int wv   = tid >> 5;
    const int lane = tid & 31;
    const int h    = lane >> 4;
    const int c    = lane & 15;
    const bool cok = (c < 9);
    const int cc   = cok ? c : 8;      // clamped column: every lane loads in-row

    const long long b = (long long)blockIdx.x * 8 + wv;
    const bool active = (b < (long long)Bn);
    const long long bb = active ? b : 0;   // clamp so inactive waves read valid mem

    const float* Ab = coords + bb * (long long)N * 9;
    const float* Fb = fest   + bb * 9;

    const float fc = cok ? Fb[c] : 0.0f;   // 0 for pad lanes -> dot stays correct

    // per-lane base: rows (2h, 2h+1), column cc
    const float* Ap = Ab + (2 * h) * 9 + cc;

    v8f acc = {0.f, 0.f, 0.f, 0.f, 0.f, 0.f, 0.f, 0.f};

    #pragma unroll 4
    for (int i = 0; i < N; i += 4) {
        // A is ~302MB read-once (> 192MB L2): non-temporal streaming loads
        const float a0 = __builtin_nontemporal_load(Ap + i * 9);
        const float a1 = __builtin_nontemporal_load(Ap + i * 9 + 9);
        // residual r = dot(A[row], f), reduced over the 16-lane half via DPP
        const float r0 = half_reduce16(a0 * fc);
        const float r1 = half_reduce16(a1 * fc);
        // scale = sqrt(w) = 1/sqrt(r^2 + delta)  (single v_rsq_f32)
        const float s0 = __builtin_amdgcn_rsqf(fmaf(r0, r0, WEIGHT_DELTA));
        const float s1 = __builtin_amdgcn_rsqf(fmaf(r1, r1, WEIGHT_DELTA));
        v2f ab;
        ab[0] = cok ? a0 * s0 : 0.0f;      // zero pad columns for the Gram
        ab[1] = cok ? a1 * s1 : 0.0f;
        // M += G_chunk^T * G_chunk  (A-operand register image == B-operand)
        acc = __builtin_amdgcn_wmma_f32_16x16x4_f32(
            /*neg_a=*/false, ab, /*neg_b=*/false, ab,
            /*c_mod=*/(short)0, acc, /*reuse_a=*/false, /*reuse_b=*/false);
    }

    // Scatter 16x16 C tile into per-wave LDS as row-major 9x9:
    // element (m = v + 8h, n = c) lives in acc[v] of this lane.
    float* Mw = &shM[wv][0];
    float* Vw = &shV[wv][0];
    if (cok) {
        #pragma unroll
        for (int v = 0; v < 8; ++v) {
            const int m = v + 8 * h;
            if (m < 9) Mw[m * 9 + c] = acc[v];
        }
    }
    __syncthreads();

    if (active) {
        // Init V = I, M += eps*I (lanes 0..8 in parallel)
        if (lane < 9) {
            #pragma unroll
            for (int j = 0; j < 9; ++j) Vw[lane * 9 + j] = (lane == j) ? 1.0f : 0.0f;
            Mw[lane * 9 + lane] += EIGH_EPS;
        }

        // Cyclic Jacobi eigensolver on 9x9 symmetric M, vectors accumulated in V.
        // Rotation params are wave-uniform (all lanes read same LDS words);
        // row/col/V updates are parallel over k = lane (lanes 0..8).
        for (int sweep = 0; sweep < JACOBI_SWEEPS; ++sweep) {
            for (int p = 0; p < 8; ++p) {
                for (int q = p + 1; q < 9; ++q) {
                    const float apq = Mw[p * 9 + q];
                    const float app = Mw[p * 9 + p];
                    const float aqq = Mw[q * 9 + q];
                    const bool rot = fabsf(apq) > 1e-26f;
                    const float tau = (aqq - app) * 0.5f * __builtin_amdgcn_rcpf(apq);
                    const float t = ((tau >= 0.0f) ? 1.0f : -1.0f) *
                                    __builtin_amdgcn_rcpf(
                                        fabsf(tau) +
                                        __builtin_amdgcn_sqrtf(fmaf(tau, tau, 1.0f)));
                    float ct = __builtin_amdgcn_rsqf(fmaf(t, t, 1.0f));
                    float st = t * ct;
                    ct = rot ? ct : 1.0f;
                    st = rot ? st : 0.0f;
                    const int k = lane;
                    if (lane < 9) {
                        // rows p,q:  J^T * A
                        const float mpk = Mw[p * 9 + k];
                        const float mqk = Mw[q * 9 + k];
                        Mw[p * 9 + k] = ct * mpk - st * mqk;
                        Mw[q * 9 + k] = st * mpk + ct * mqk;
                    }
                    if (lane < 9) {
                        // cols p,q:  (J^T A) * J   (in-wave LDS ops are in-order)
                        const float mkp = Mw[k * 9 + p];
                        const float mkq = Mw[k * 9 + q];
                        Mw[k * 9 + p] = ct * mkp - st * mkq;
                        Mw[k * 9 + q] = st * mkp + ct * mkq;
                        const float vkp = Vw[k * 9 + p];
                        const float vkq = Vw[k * 9 + q];
                        Vw[k * 9 + p] = ct * vkp - st * vkq;
                        Vw[k * 9 + q] = st * vkp + ct * vkq;
                    }
                }
            }
        }

        if (lane == 0) {
            // smallest eigenvalue -> its eigenvector column
            int jmin = 0;
            float emin = Mw[0];
            for (int j = 1; j < 9; ++j) {
                const float e = Mw[j * 9 + j];
                if (e < emin) { emin = e; jmin = j; }
            }
            // deterministic sign: largest-|component| made positive
            float best = 0.0f;
            for (int k = 0; k < 9; ++k) {
                const float v = Vw[k * 9 + jmin];
                if (fabsf(v) > fabsf(best)) best = v;
            }
            const float sgn = (best < 0.0f) ? -1.0f : 1.0f;
            float* op = out + b * 9;
            for (int k = 0; k < 9; ++k) op[k] = sgn * Vw[k * 9 + jmin];
        }
    }
}

extern "C" void kernel_launch(void* const* d_in, const int* in_sizes, int n_in,
                              void* d_out, int out_size, void* d_ws, size_t ws_size,
                              hipStream_t stream) {
    const float* coords = (const float*)d_in[0];  // [B,1024,9] f32
    const float* fest   = (const float*)d_in[1];  // [B,9]      f32
    // d_in[2] is call_id (scalar int); reference takes the call_id < 0 path.
    float* out = (float*)d_out;                   // [B,9]      f32

    const int Bn = in_sizes[1] / 9;
    const int N  = in_sizes[0] / (Bn * 9);        // 1024, multiple of 4

    const int blocks = (Bn + 7) / 8;              // 8 waves (batches) per block
    irls_step_kernel<<<blocks, 256, 0, stream>>>(coords, fest, out, Bn, N);
}